// OTLoss_29875792511744
// MI455X (gfx1250) — compile-verified
//
#include <hip/hip_runtime.h>
#include <math.h>

// ---------------------------------------------------------------------------
// Debiased Sinkhorn divergence (geomloss defaults: blur=0.05, scaling=0.5, p=1)
// on B=2, N=M=4096, D=2 point clouds, for MI455X (gfx1250, wave32, WMMA).
//
// Cost tile trick:  d2_ij = [qx qy |q|^2 1] . [-2px -2py 1 |p|^2]
// => one V_WMMA_F32_16X16X4_F32 yields a full 16x16 tile of squared distances.
// M-dim = j (cost columns), N-dim = i (output rows): each lane then owns 8
// d2 values of ONE row i -> lane-local streaming logsumexp, one shfl_xor(16)
// merge at the end.  Transcendentals use the raw v_sqrt_f32 / v_exp_f32 /
// v_log_f32 hardware ops (no ocml denorm/ulp fixup tails).
// ---------------------------------------------------------------------------

typedef float v2f __attribute__((ext_vector_type(2)));
typedef float v8f __attribute__((ext_vector_type(8)));

#define NPTS 4096
#define BATCH 2
#define MAX_IT 40          // even: after padding, state parity returns to 0
#define ROWS_PER_BLK 128   // 8 waves x 16 rows
#define THREADS 256
#define BLUR 0.05f

// workspace layout (floats):
//   [0 .. 63]                       eps array: [0]=init eps (=diameter),
//                                   [1..MAX_IT]=loop eps (0 => padded copy)
//   S  = ws+64 : state duals  [dual(4)][parity(2)][batch(2)][NPTS]
//   F  = S + 4*2*BATCH*NPTS  : final duals [dual(4)][batch(2)][NPTS]
#define WS_S_OFF 64
#define WS_F_OFF (WS_S_OFF + 4 * 2 * BATCH * NPTS)

// dual indices: 0=f_ba (over x), 1=g_ab (over y), 2=f_aa (x), 3=g_bb (y)

__device__ __forceinline__ float fast_sqrt(float x) { return __builtin_amdgcn_sqrtf(x); }
__device__ __forceinline__ float fast_exp2(float x) { return __builtin_amdgcn_exp2f(x); }
__device__ __forceinline__ float fast_log2(float x) { return __builtin_amdgcn_logf(x); }

// ---------------------------------------------------------------------------
// Kernel 1: diameter + epsilon schedule (geomloss epsilon_schedule, P=1)
// ---------------------------------------------------------------------------
__global__ __launch_bounds__(THREADS) void ot_sched(const float* __restrict__ pre,
                                                    const float* __restrict__ gt,
                                                    float* __restrict__ ws) {
  __shared__ float red[4][THREADS];
  const int tid = threadIdx.x;
  const int TOT = BATCH * NPTS;  // points per tensor
  float mnx = INFINITY, mxx = -INFINITY, mny = INFINITY, mxy = -INFINITY;
  for (int k = tid; k < 2 * TOT; k += THREADS) {
    const float* src = (k < TOT) ? pre : gt;
    const int p = (k < TOT) ? k : (k - TOT);
    const float x = src[2 * p + 0];
    const float y = src[2 * p + 1];
    mnx = fminf(mnx, x); mxx = fmaxf(mxx, x);
    mny = fminf(mny, y); mxy = fmaxf(mxy, y);
  }
  red[0][tid] = mnx; red[1][tid] = mxx; red[2][tid] = mny; red[3][tid] = mxy;
  __syncthreads();
  for (int off = THREADS / 2; off > 0; off >>= 1) {
    if (tid < off) {
      red[0][tid] = fminf(red[0][tid], red[0][tid + off]);
      red[1][tid] = fmaxf(red[1][tid], red[1][tid + off]);
      red[2][tid] = fminf(red[2][tid], red[2][tid + off]);
      red[3][tid] = fmaxf(red[3][tid], red[3][tid + off]);
    }
    __syncthreads();
  }
  if (tid == 0) {
    const double dx = (double)red[1][0] - (double)red[0][0];
    const double dy = (double)red[3][0] - (double)red[2][0];
    const double diam = sqrt(dx * dx + dy * dy);
    ws[0] = (float)diam;             // init eps = eps_list[0]
    int idx = 1;
    ws[idx++] = (float)diam;         // loop iter 0: eps_list[0]
    // arange(log d, log blur, log 0.5) -> exp()
    double t = log(diam);
    const double stop = log((double)BLUR);
    const double step = log(0.5);
    while (t > stop && idx < MAX_IT) { ws[idx++] = (float)exp(t); t += step; }
    if (idx <= MAX_IT) ws[idx++] = BLUR;     // eps_list last element
    while (idx <= MAX_IT) ws[idx++] = 0.0f;  // padding -> copy-through phases
  }
}

// ---------------------------------------------------------------------------
// Kernel 2: one "phase" = 4 softmins x 2 batches.
//   phase 0           : init   (h = base log only, raw write, parity 0)
//   phase 1..MAX_IT   : loop t (h = base + dual/eps, out = 0.5*(old+new),
//                                ping-pong parity; eps==0 -> copy)
//   phase MAX_IT+1    : final  (raw write to F, eps = blur)
// ---------------------------------------------------------------------------
__global__ __launch_bounds__(THREADS) void ot_phase(const float* __restrict__ pre,
                                                    const float* __restrict__ gt,
                                                    float* __restrict__ ws,
                                                    int phase) {
  float* S = ws + WS_S_OFF;
  float* F = ws + WS_F_OFF;

  const int task = blockIdx.y & 3;   // 0:f_ba 1:g_ab 2:f_aa 3:g_bb
  const int b    = blockIdx.y >> 2;

  // task geometry: row cloud, col cloud, dual added to h (over cols), output dual
  const int rowc[4] = {0, 1, 0, 1};
  const int colc[4] = {1, 0, 0, 1};
  const int hd[4]   = {1, 0, 2, 3};
  const int od[4]   = {0, 1, 2, 3};

  const bool is_init  = (phase == 0);
  const bool is_final = (phase > MAX_IT);
  const int t  = phase - 1;
  const int rp = (is_init || is_final) ? 0 : (t & 1);
  const int wp = is_init ? 0 : ((t + 1) & 1);

  float eps;
  if (is_init)       eps = ws[0];
  else if (is_final) eps = BLUR;
  else               eps = ws[phase];

  const int tid  = threadIdx.x;
  const int row0 = blockIdx.x * ROWS_PER_BLK;

  float*       outS = &S[((od[task] * 2 + wp) * BATCH + b) * NPTS];
  const float* inS  = &S[((od[task] * 2 + rp) * BATCH + b) * NPTS];

  if (!is_init && !is_final && eps == 0.0f) {
    // padded iteration: preserve ping-pong parity with a copy
    if (tid < ROWS_PER_BLK) outS[row0 + tid] = inS[row0 + tid];
    return;
  }

  __shared__ float qx[NPTS], qy[NPTS], hh[NPTS];

  const float inv_eps  = 1.0f / eps;
  const float LOG2E    = 1.4426950408889634f;
  const float LN2      = 0.6931471805599453f;
  const float ivl2     = inv_eps * LOG2E;          // C/eps in log2 domain
  const float base_log = -logf((float)NPTS);       // a_log == b_log (N==M)

  const float* Q    = (colc[task] == 0 ? pre : gt) + (size_t)b * NPTS * 2;
  const float* P    = (rowc[task] == 0 ? pre : gt) + (size_t)b * NPTS * 2;
  const float* dual = &S[((hd[task] * 2 + rp) * BATCH + b) * NPTS];

  // stage column cloud + h vector (pre-scaled by log2 e) into LDS
  for (int idx = tid; idx < NPTS; idx += THREADS) {
    qx[idx] = Q[2 * idx + 0];
    qy[idx] = Q[2 * idx + 1];
    float hv = base_log;
    if (!is_init) hv += dual[idx] * inv_eps;
    hh[idx] = hv * LOG2E;
  }
  __syncthreads();

  const int wave = tid >> 5;
  const int lane = tid & 31;
  const int il   = lane & 15;
  const int hi   = lane >> 4;
  const int i    = row0 + wave * 16 + il;

  // B operand (4x16, K x N=i), constant over the whole j loop.
  // lanes 0-15: K=0 row (-2px), K=1 row (-2py); lanes 16-31: K=2 (1), K=3 (|p|^2)
  const float px = P[2 * i + 0];
  const float py = P[2 * i + 1];
  v2f Bv;
  Bv.x = hi ? 1.0f : (-2.0f * px);
  Bv.y = hi ? (px * px + py * py) : (-2.0f * py);

  float m = -INFINITY;   // running max (log2 domain)
  float s = 0.0f;        // running sum of exp2(v - m)

#pragma unroll 2
  for (int jb = 0; jb < NPTS; jb += 16) {
    // A operand (16x4, M=j x K): lanes 0-15: K0=qx,K1=qy; lanes 16-31: K2=|q|^2,K3=1
    const int jj = jb + il;
    const float ax = qx[jj];
    const float ay = qy[jj];
    v2f Av;
    Av.x = hi ? (ax * ax + ay * ay) : ax;
    Av.y = hi ? 1.0f : ay;

    v8f c = {};
    v8f d2 = __builtin_amdgcn_wmma_f32_16x16x4_f32(
        /*neg_a=*/false, Av, /*neg_b=*/false, Bv,
        /*c_mod=*/(short)0, c, /*reuse_a=*/false, /*reuse_b=*/false);

    // C/D layout: VGPR r, this lane -> element (j = jb + r + 8*hi, i = row fixed)
    const float* hp = &hh[jb + hi * 8];
    float v[8];
#pragma unroll
    for (int r = 0; r < 8; ++r) {
      const float c_ij = fast_sqrt(fmaxf(d2[r], 1e-12f));
      v[r] = hp[r] - c_ij * ivl2;   // (h_j - C_ij/eps) * log2(e)
    }
    // shallow max tree (dual-issue friendly)
    const float m01 = fmaxf(v[0], v[1]);
    const float m23 = fmaxf(v[2], v[3]);
    const float m45 = fmaxf(v[4], v[5]);
    const float m67 = fmaxf(v[6], v[7]);
    const float tm  = fmaxf(fmaxf(m01, m23), fmaxf(m45, m67));
    if (tm > m) { s *= fast_exp2(m - tm); m = tm; }
#pragma unroll
    for (int r = 0; r < 8; ++r) s += fast_exp2(v[r] - m);
  }

  // merge the two j-halves held by lane and lane^16 (same row i)
  const float om = __shfl_xor(m, 16, 32);
  const float os = __shfl_xor(s, 16, 32);
  const float M  = fmaxf(m, om);
  const float st = s * fast_exp2(m - M) + os * fast_exp2(om - M);
  const float f  = -eps * LN2 * (M + fast_log2(st));  // -eps * logsumexp (nat. log)

  if (hi == 0) {
    if (is_final)      F[(od[task] * BATCH + b) * NPTS + i] = f;
    else if (is_init)  outS[i] = f;
    else               outS[i] = 0.5f * (inS[i] + f);  // symmetric averaging
  }
}

// ---------------------------------------------------------------------------
// Kernel 3: S = mean_b [ <a, f_ba - f_aa> + <b, g_ab - g_bb> ],  a=b=1/N
// ---------------------------------------------------------------------------
__global__ __launch_bounds__(THREADS) void ot_reduce(const float* __restrict__ ws,
                                                     float* __restrict__ out) {
  const float* F = ws + WS_F_OFF;
  __shared__ float red[THREADS];
  const int tid = threadIdx.x;
  float acc = 0.0f;
  for (int b = 0; b < BATCH; ++b) {
    const float* fba = &F[(0 * BATCH + b) * NPTS];
    const float* gab = &F[(1 * BATCH + b) * NPTS];
    const float* faa = &F[(2 * BATCH + b) * NPTS];
    const float* gbb = &F[(3 * BATCH + b) * NPTS];
    for (int i = tid; i < NPTS; i += THREADS)
      acc += (fba[i] - faa[i]) + (gab[i] - gbb[i]);
  }
  red[tid] = acc;
  __syncthreads();
  for (int off = THREADS / 2; off > 0; off >>= 1) {
    if (tid < off) red[tid] += red[tid + off];
    __syncthreads();
  }
  if (tid == 0) out[0] = red[0] / ((float)NPTS * (float)BATCH);
}

// ---------------------------------------------------------------------------
extern "C" void kernel_launch(void* const* d_in, const int* in_sizes, int n_in,
                              void* d_out, int out_size, void* d_ws, size_t ws_size,
                              hipStream_t stream) {
  (void)in_sizes; (void)n_in; (void)out_size; (void)ws_size;
  const float* pre = (const float*)d_in[0];
  const float* gt  = (const float*)d_in[1];
  float* ws  = (float*)d_ws;
  float* out = (float*)d_out;

  ot_sched<<<1, THREADS, 0, stream>>>(pre, gt, ws);

  // phase 0 = init, 1..MAX_IT = annealing loop (padded), MAX_IT+1 = final
  for (int phase = 0; phase <= MAX_IT + 1; ++phase) {
    ot_phase<<<dim3(NPTS / ROWS_PER_BLK, 4 * BATCH), THREADS, 0, stream>>>(
        pre, gt, ws, phase);
  }

  ot_reduce<<<1, THREADS, 0, stream>>>(ws, out);
}